// REN_56066503082549
// MI455X (gfx1250) — compile-verified
//
#include <hip/hip_runtime.h>
#include <math.h>

#define VSZ  50000
#define EDIM 128
#define KMEM 20
#define BB   64
#define TT   100
#define SS   30

typedef float v2f __attribute__((ext_vector_type(2)));
typedef float v8f __attribute__((ext_vector_type(8)));

static __device__ __forceinline__ v8f v8f_zero() {
    v8f r;
#pragma unroll
    for (int i = 0; i < 8; ++i) r[i] = 0.0f;
    return r;
}

// D = A(16x4 f32) * B(4x16 f32) + C  -- CDNA5 V_WMMA_F32_16X16X4_F32
static __device__ __forceinline__ v8f wmma4_f32(v2f a, v2f b, v8f c) {
    return __builtin_amdgcn_wmma_f32_16x16x4_f32(false, a, false, b, (short)0, c, false, false);
}

// ---------------------------------------------------------------------------
// 1) x_enc[t][b][:] = sum_s embed[x[b][t][s]][:]      (layout [T,B,E])
// ---------------------------------------------------------------------------
__global__ void k_embed_x(const int* __restrict__ x, const float* __restrict__ embed,
                          float* __restrict__ xe) {
    const int bid = blockIdx.x;         // T*B blocks
    const int t = bid / BB, b = bid % BB;
    const int e = threadIdx.x;          // 128 threads
    const int* xp = x + (b * TT + t) * SS;
    float acc = 0.0f;
#pragma unroll 5
    for (int s = 0; s < SS; ++s) acc += embed[xp[s] * EDIM + e];
    xe[(t * BB + b) * EDIM + e] = acc;
}

// ---------------------------------------------------------------------------
// 2) q[b][:] = sum_s embed[query[b][s]][:]
// ---------------------------------------------------------------------------
__global__ void k_embed_q(const int* __restrict__ query, const float* __restrict__ embed,
                          float* __restrict__ q) {
    const int b = blockIdx.x;
    const int e = threadIdx.x;
    const int* qp = query + b * SS;
    float acc = 0.0f;
#pragma unroll 5
    for (int s = 0; s < SS; ++s) acc += embed[qp[s] * EDIM + e];
    q[b * EDIM + e] = acc;
}

// ---------------------------------------------------------------------------
// 3) XM[m][k] = dot(xe_row m, w_mem[k])   (m = t*B+b)   [scalar, tiny]
// ---------------------------------------------------------------------------
__global__ void k_xm(const float* __restrict__ xe, const float* __restrict__ w_mem,
                     float* __restrict__ xm) {
    const int gid = blockIdx.x * blockDim.x + threadIdx.x;
    if (gid >= TT * BB * KMEM) return;
    const int m = gid / KMEM, k = gid % KMEM;
    const float* a = xe + m * EDIM;
    const float* w = w_mem + k * EDIM;
    float s = 0.0f;
#pragma unroll 8
    for (int e = 0; e < EDIM; ++e) s += a[e] * w[e];
    xm[gid] = s;
}

// ---------------------------------------------------------------------------
// 4) v_term[k][e] = dot(w_mem[k], Vw[e])   [scalar, tiny]
// ---------------------------------------------------------------------------
__global__ void k_vterm(const float* __restrict__ w_mem, const float* __restrict__ Vw,
                        float* __restrict__ vt) {
    const int gid = blockIdx.x * blockDim.x + threadIdx.x;
    if (gid >= KMEM * EDIM) return;
    const int k = gid / EDIM, e = gid % EDIM;
    const float* a = w_mem + k * EDIM;
    const float* v = Vw + e * EDIM;
    float s = 0.0f;
#pragma unroll 8
    for (int j = 0; j < EDIM; ++j) s += a[j] * v[j];
    vt[gid] = s;
}

// ---------------------------------------------------------------------------
// 5) XW = x_enc @ Ww^T   : [6400,128] x [128,128], WMMA f32 16x16x4
//    grid 50 blocks x 256 thr ; each wave owns one 16-row M-tile, all 8 N-tiles
// ---------------------------------------------------------------------------
__global__ void __launch_bounds__(256) k_xw(const float* __restrict__ xe,
                                            const float* __restrict__ Ww,
                                            float* __restrict__ xw) {
    __shared__ float WwT[EDIM * 132];   // WwT[e][f] = Ww[f][e]
    const int tid = threadIdx.x;
    for (int i = tid; i < EDIM * EDIM; i += 256) {
        const int f = i >> 7, e = i & 127;
        WwT[e * 132 + f] = Ww[i];
    }
    __syncthreads();

    const int wave = tid >> 5, lane = tid & 31;
    const int lm = lane & 15, lh = lane >> 4;
    const int row0 = (blockIdx.x * 8 + wave) * 16;      // 400 M-tiles total

    v8f acc[8];
#pragma unroll
    for (int nt = 0; nt < 8; ++nt) acc[nt] = v8f_zero();

    for (int kk = 0; kk < EDIM; kk += 4) {
        v2f a;
        const int r = row0 + lm, e0 = kk + lh * 2;
        a.x = xe[r * EDIM + e0];
        a.y = xe[r * EDIM + e0 + 1];
#pragma unroll
        for (int nt = 0; nt < 8; ++nt) {
            v2f bf;
            const int n = nt * 16 + lm;
            bf.x = WwT[e0 * 132 + n];
            bf.y = WwT[(e0 + 1) * 132 + n];
            acc[nt] = wmma4_f32(a, bf, acc[nt]);
        }
    }
#pragma unroll
    for (int nt = 0; nt < 8; ++nt) {
        const int col = nt * 16 + lm;
#pragma unroll
        for (int j = 0; j < 8; ++j) {
            const int row = row0 + j + lh * 8;
            xw[row * EDIM + col] = acc[nt][j];
        }
    }
}

// ---------------------------------------------------------------------------
// 6) Recurrence: one workgroup per k (20 WGs); h[k] lives in LDS for all T.
//    Per step: gate = sigmoid(xt.h + XM), cand = prelu(h@U^T + vt + XW + bias),
//    h += gate*cand, h /= (rowsum + 1e-8).  GEMM via WMMA f32 16x16x4.
// ---------------------------------------------------------------------------
__global__ void __launch_bounds__(256) k_recur(const float* __restrict__ xe,
                                               const float* __restrict__ xw,
                                               const float* __restrict__ xm,
                                               const float* __restrict__ vt,
                                               const float* __restrict__ bias,
                                               const float* __restrict__ prelu_mem,
                                               const float* __restrict__ U,
                                               float* __restrict__ hg) {
    __shared__ float Ut[EDIM * 132];    // Ut[e][f] = U[f][e]  (B matrix)
    __shared__ float hS[BB * 132];      // h[k] tile, rows b, cols e
    __shared__ float gred[BB * 4];
    __shared__ float gateS[BB];
    __shared__ float rsum[BB];

    const int tid = threadIdx.x;
    const int k = blockIdx.x;
    const int wave = tid >> 5, lane = tid & 31;
    const int lm = lane & 15, lh = lane >> 4;
    const int mtile = wave >> 1;               // 4 M-tiles (b bands of 16)
    const int ntbase = (wave & 1) * 4;         // N-tiles 0..3 or 4..7
    const int rb = tid >> 2, part = tid & 3;   // per-row reduction mapping

    for (int i = tid; i < EDIM * EDIM; i += 256) {
        const int f = i >> 7, e = i & 127;
        Ut[e * 132 + f] = U[i];
    }
    for (int i = tid; i < BB * 132; i += 256) hS[i] = 0.0f;
    __syncthreads();

    for (int t = 0; t < TT; ++t) {
        const float* xt  = xe + t * BB * EDIM;
        const float* xwt = xw + t * BB * EDIM;
        const float* xmt = xm + t * BB * KMEM;

        // gate partial dots (read-only on hS)
        {
            float p = 0.0f;
            const int e0 = part * 32;
#pragma unroll 8
            for (int j = 0; j < 32; ++j) p += xt[rb * EDIM + e0 + j] * hS[rb * 132 + e0 + j];
            gred[rb * 4 + part] = p;
        }

        // cand_pre = h @ U^T  (read-only on hS)
        v8f acc[4];
#pragma unroll
        for (int nt = 0; nt < 4; ++nt) acc[nt] = v8f_zero();
        for (int kk = 0; kk < EDIM; kk += 4) {
            v2f a;
            const int r = mtile * 16 + lm, e0 = kk + lh * 2;
            a.x = hS[r * 132 + e0];
            a.y = hS[r * 132 + e0 + 1];
#pragma unroll
            for (int nt = 0; nt < 4; ++nt) {
                v2f bf;
                const int n = (ntbase + nt) * 16 + lm;
                bf.x = Ut[e0 * 132 + n];
                bf.y = Ut[(e0 + 1) * 132 + n];
                acc[nt] = wmma4_f32(a, bf, acc[nt]);
            }
        }
        __syncthreads();   // all hS reads + gred writes done

        if (tid < BB) {
            const float g = gred[tid * 4] + gred[tid * 4 + 1] + gred[tid * 4 + 2] +
                            gred[tid * 4 + 3] + xmt[tid * KMEM + k];
            gateS[tid] = 1.0f / (1.0f + expf(-g));
        }
        __syncthreads();

        // epilogue: cand = prelu(.), h += gate * cand
#pragma unroll
        for (int nt = 0; nt < 4; ++nt) {
            const int col = (ntbase + nt) * 16 + lm;
            const float vtc = vt[k * EDIM + col];
            const float bc  = bias[col];
            const float pmc = prelu_mem[col];
#pragma unroll
            for (int j = 0; j < 8; ++j) {
                const int row = mtile * 16 + j + lh * 8;
                float cp = acc[nt][j] + vtc + xwt[row * EDIM + col] + bc;
                cp = (cp >= 0.0f) ? cp : pmc * cp;
                hS[row * 132 + col] += gateS[row] * cp;
            }
        }
        __syncthreads();

        // row sums and normalization
        {
            float p = 0.0f;
            const int e0 = part * 32;
#pragma unroll 8
            for (int j = 0; j < 32; ++j) p += hS[rb * 132 + e0 + j];
            gred[rb * 4 + part] = p;
        }
        __syncthreads();
        if (tid < BB)
            rsum[tid] = gred[tid * 4] + gred[tid * 4 + 1] + gred[tid * 4 + 2] +
                        gred[tid * 4 + 3] + 1e-8f;
        __syncthreads();
        {
            const float inv = rsum[rb];
            const int e0 = part * 32;
#pragma unroll 8
            for (int j = 0; j < 32; ++j) hS[rb * 132 + e0 + j] /= inv;
        }
        __syncthreads();
    }

    // write final h[k] to global: [K,B,E]
    for (int i = tid; i < BB * EDIM; i += 256) {
        const int b = i >> 7, e = i & 127;
        hg[k * BB * EDIM + i] = hS[b * 132 + e];
    }
}

// ---------------------------------------------------------------------------
// 7) Attention + mix + z = prelu(q + u@H^T), stored transposed zT[e][b].
//    Single workgroup, scalar (tiny: ~2 MFLOP).
// ---------------------------------------------------------------------------
__global__ void __launch_bounds__(256) k_att_uz(const float* __restrict__ hg,
                                                const float* __restrict__ q,
                                                const float* __restrict__ H,
                                                const float* __restrict__ prelu_out,
                                                float* __restrict__ zT) {
    __shared__ float attS[KMEM * BB];
    __shared__ float uS[BB * EDIM];
    const int tid = threadIdx.x;

    // att[k][b] = dot(h[k][b], q[b])
    for (int i = tid; i < KMEM * BB; i += 256) {
        const int k = i / BB, b = i % BB;
        const float* hp = hg + (k * BB + b) * EDIM;
        const float* qp = q + b * EDIM;
        float s = 0.0f;
#pragma unroll 8
        for (int e = 0; e < EDIM; ++e) s += hp[e] * qp[e];
        attS[i] = s;
    }
    __syncthreads();

    // softmax over b (axis=1, faithful to reference quirk)
    if (tid < KMEM) {
        float m = -3.4e38f;
        for (int b = 0; b < BB; ++b) m = fmaxf(m, attS[tid * BB + b]);
        float s = 0.0f;
        for (int b = 0; b < BB; ++b) {
            const float e = expf(attS[tid * BB + b] - m);
            attS[tid * BB + b] = e;
            s += e;
        }
        const float inv = 1.0f / s;
        for (int b = 0; b < BB; ++b) attS[tid * BB + b] *= inv;
    }
    __syncthreads();

    // u[b][e] = sum_k att[k][b] * h[k][b][e]
    for (int i = tid; i < BB * EDIM; i += 256) {
        const int b = i >> 7, e = i & 127;
        float s = 0.0f;
#pragma unroll
        for (int k = 0; k < KMEM; ++k)
            s += attS[k * BB + b] * hg[(k * BB + b) * EDIM + e];
        uS[i] = s;
    }
    __syncthreads();

    // z[b][e] = prelu(q[b][e] + dot(u[b][:], H[e][:]), prelu_out); store zT[e][b]
    const float po = prelu_out[0];
    for (int i = tid; i < BB * EDIM; i += 256) {
        const int b = i >> 7, e = i & 127;
        float s = q[b * EDIM + e];
        const float* hr = H + e * EDIM;
        const float* ur = uS + b * EDIM;
#pragma unroll 8
        for (int e2 = 0; e2 < EDIM; ++e2) s += ur[e2] * hr[e2];
        s = (s >= 0.0f) ? s : po * s;
        zT[e * BB + b] = s;
    }
}

// ---------------------------------------------------------------------------
// 8) out[b][v] = sum_e z[b][e] * R[v][e].  Computed as out^T = R @ zT via
//    WMMA (R rows are row-major A loads; zT rows are the B matrix in LDS).
//    V = 50000 = 3125 exact 16-row tiles. Each wave: one v-tile x 4 b-tiles.
// ---------------------------------------------------------------------------
__global__ void __launch_bounds__(256) k_out(const float* __restrict__ R,
                                             const float* __restrict__ zT,
                                             float* __restrict__ out) {
    __shared__ float zTS[EDIM * 68];    // zTS[e][b]
    const int tid = threadIdx.x;
    for (int i = tid; i < EDIM * BB; i += 256) {
        const int e = i >> 6, b = i & 63;
        zTS[e * 68 + b] = zT[i];
    }
    __syncthreads();

    const int wave = tid >> 5, lane = tid & 31;
    const int vtile = blockIdx.x * 8 + wave;
    if (vtile >= VSZ / 16) return;      // after barrier; safe
    const int lm = lane & 15, lh = lane >> 4;
    const int v0 = vtile * 16;

    v8f acc[4];
#pragma unroll
    for (int nt = 0; nt < 4; ++nt) acc[nt] = v8f_zero();

    for (int kk = 0; kk < EDIM; kk += 4) {
        v2f a;
        const int e0 = kk + lh * 2;
        a.x = R[(v0 + lm) * EDIM + e0];
        a.y = R[(v0 + lm) * EDIM + e0 + 1];
#pragma unroll
        for (int nt = 0; nt < 4; ++nt) {
            v2f bf;
            const int n = nt * 16 + lm;
            bf.x = zTS[e0 * 68 + n];
            bf.y = zTS[(e0 + 1) * 68 + n];
            acc[nt] = wmma4_f32(a, bf, acc[nt]);
        }
    }

    // lane holds rows v0 + 8*lh + (0..7) for column b = nt*16+lm : contiguous,
    // 16B-aligned (50000 % 4 == 0) -> two float4 stores per tile.
#pragma unroll
    for (int nt = 0; nt < 4; ++nt) {
        const int bcol = nt * 16 + lm;
        float4* p = (float4*)(out + (size_t)bcol * VSZ + v0 + 8 * lh);
        float4 lo, hi;
        lo.x = acc[nt][0]; lo.y = acc[nt][1]; lo.z = acc[nt][2]; lo.w = acc[nt][3];
        hi.x = acc[nt][4]; hi.y = acc[nt][5]; hi.z = acc[nt][6]; hi.w = acc[nt][7];
        p[0] = lo;
        p[1] = hi;
    }
}

// ---------------------------------------------------------------------------
extern "C" void kernel_launch(void* const* d_in, const int* in_sizes, int n_in,
                              void* d_out, int out_size, void* d_ws, size_t ws_size,
                              hipStream_t stream) {
    const int*   x        = (const int*)d_in[0];
    const int*   query    = (const int*)d_in[1];
    const float* embed    = (const float*)d_in[2];
    const float* w_mem    = (const float*)d_in[3];
    const float* U        = (const float*)d_in[4];
    const float* Vw       = (const float*)d_in[5];
    const float* Ww       = (const float*)d_in[6];
    const float* bias     = (const float*)d_in[7];
    const float* pmem     = (const float*)d_in[8];
    const float* R        = (const float*)d_in[9];
    const float* H        = (const float*)d_in[10];
    const float* pout     = (const float*)d_in[11];
    float* out = (float*)d_out;

    float* W = (float*)d_ws;
    float* xe = W;                                   // T*B*E  = 819200
    float* xw = xe + TT * BB * EDIM;                 // 819200
    float* xm = xw + TT * BB * EDIM;                 // T*B*K  = 128000
    float* qv = xm + TT * BB * KMEM;                 // B*E    = 8192
    float* vt = qv + BB * EDIM;                      // K*E    = 2560
    float* hg = vt + KMEM * EDIM;                    // K*B*E  = 163840
    float* zT = hg + KMEM * BB * EDIM;               // E*B    = 8192

    k_embed_x<<<TT * BB, EDIM, 0, stream>>>(x, embed, xe);
    k_embed_q<<<BB, EDIM, 0, stream>>>(query, embed, qv);
    k_xm<<<(TT * BB * KMEM + 255) / 256, 256, 0, stream>>>(xe, w_mem, xm);
    k_vterm<<<(KMEM * EDIM + 255) / 256, 256, 0, stream>>>(w_mem, Vw, vt);
    k_xw<<<50, 256, 0, stream>>>(xe, Ww, xw);
    k_recur<<<KMEM, 256, 0, stream>>>(xe, xw, xm, vt, bias, pmem, U, hg);
    k_att_uz<<<1, 256, 0, stream>>>(hg, qv, H, pout, zT);
    k_out<<<(VSZ / 16 + 7) / 8, 256, 0, stream>>>(R, zT, out);
}